// PointNetEncoder_57217554317539
// MI455X (gfx1250) — compile-verified
//
#include <hip/hip_runtime.h>
#include <math.h>

// ---------------- problem constants ----------------
#define WIDTH   224
#define BATCH   64
#define NPTS    (BATCH * WIDTH * WIDTH)     // 3,211,264
#define KFPS    512
#define LN_EPS  1e-5f

// FPS scan geometry: 392 blocks * 256 threads * 32 pts = NPTS exactly
#define U_THREADS 256
#define U_PPT     32
#define U_CHUNK   (U_THREADS * U_PPT)       // 8192
#define U_BLOCKS  (NPTS / U_CHUNK)          // 392

// ---------------- workspace layout (float offsets) ----------------
#define OFF_PX    0
#define OFF_PY    (NPTS)
#define OFF_PZ    (2 * NPTS)
#define OFF_MIND  (3 * NPTS)
#define OFF_CTRL  (4 * NPTS)                 // int: first-valid index
#define OFF_PARTA (4 * NPTS + 16)            // 512 u64 partials (ping)
#define OFF_PARTB (OFF_PARTA + 1024)         // 512 u64 partials (pong)
#define OFF_IDXS  (OFF_PARTB + 1024)         // 512 ints
#define OFF_FEAT  (OFF_IDXS + 512)           // [512][8] padded features
#define OFF_W1P   (OFF_FEAT + 4096)          // [8][64] zero-padded W1
#define OFF_H1    (OFF_W1P + 512)            // [512][64]
#define OFF_H2    (OFF_H1 + 512 * 64)        // [512][128]
#define OFF_H3    (OFF_H2 + 512 * 128)       // [512][256]
#define OFF_H4    (OFF_H3 + 512 * 256)       // [512][512]
#define OFF_POOL  (OFF_H4 + 512 * 512)       // [512]
// total ~13.35M floats ~= 53.4 MB

typedef __attribute__((ext_vector_type(2))) float v2f;
typedef __attribute__((ext_vector_type(8))) float v8f;

// pack (value, index) so that u64-max == (max value, min index on ties)
__device__ __forceinline__ unsigned long long packKey(float v, int idx) {
    unsigned int u = __float_as_uint(v);
    u = (u & 0x80000000u) ? ~u : (u | 0x80000000u);
    return ((unsigned long long)u << 32) | (unsigned int)(~idx);
}
__device__ __forceinline__ unsigned long long umax64(unsigned long long a,
                                                     unsigned long long b) {
    return a > b ? a : b;
}

// ---------------- kernels ----------------
__global__ void kinit(int* ctrl) { ctrl[0] = 0x7FFFFFFF; }

__global__ void kpoints(const float* __restrict__ x,
                        float* __restrict__ px, float* __restrict__ py,
                        float* __restrict__ pz, float* __restrict__ mind,
                        int* __restrict__ ctrl) {
    int n = blockIdx.x * 256 + threadIdx.x;          // exact coverage of NPTS
    const float* xp = x + (size_t)n * 5;
    float Z = xp[3], seg = xp[4];
    int rem = n % (WIDTH * WIDTH);
    int row = rem / WIDTH, col = rem % WIDTH;
    const float FX = 224.0f * 50.0f / 20.995f;
    float ug = -((float)col - 112.0f) / FX;
    float vg = -((float)row - 112.0f) / FX;
    float X = ug * Z, Y = vg * Z;
    const float C = 0.8386705679454240f, S = 0.5446390350150271f;  // 33 deg
    px[n] = X;
    py[n] = C * Y - S * Z + 1.5f;
    pz[n] = S * Y + C * Z - 2.5f;
    bool valid = (Z < 3.0f) && (seg != 15.0f);
    mind[n] = valid ? __builtin_inff() : -__builtin_inff();
    if (valid) atomicMin(ctrl, n);
}

__global__ void kseed(const int* __restrict__ ctrl,
                      unsigned long long* __restrict__ partA) {
    int t = threadIdx.x;
    int start = ctrl[0];
    if (start == 0x7FFFFFFF) start = 0;
    if (t < U_BLOCKS)
        partA[t] = (t == 0) ? packKey(__builtin_inff(), start)
                            : packKey(-__builtin_inff(), 0);
}

// one FPS iteration: global argmax of previous partials (redundant per block),
// then distance update of this block's chunk + new per-block partial.
__global__ __launch_bounds__(U_THREADS) void kfps(
        const float* __restrict__ px, const float* __restrict__ py,
        const float* __restrict__ pz, float* __restrict__ mind,
        const unsigned long long* __restrict__ pin,
        unsigned long long* __restrict__ pout,
        int* __restrict__ idxs, int k) {
    __shared__ unsigned long long sdata[U_THREADS];
    int tid = threadIdx.x;

    // ---- prologue: reduce previous partials -> selected index ----
    unsigned long long lm = 0ull;
    for (int p = tid; p < U_BLOCKS; p += U_THREADS) lm = umax64(lm, pin[p]);
    sdata[tid] = lm;
    __syncthreads();
    for (int s = U_THREADS / 2; s > 0; s >>= 1) {
        if (tid < s) sdata[tid] = umax64(sdata[tid], sdata[tid + s]);
        __syncthreads();
    }
    int sel = (int)(~(unsigned int)sdata[0]);
    if (blockIdx.x == 0 && tid == 0) idxs[k] = sel;
    __syncthreads();   // sdata reused below

    // ---- body: distance update + local argmax of new mind ----
    float sx = px[sel], sy = py[sel], sz = pz[sel];
    unsigned long long lbest = 0ull;
    int base = blockIdx.x * U_CHUNK + tid;
#pragma unroll
    for (int j = 0; j < U_PPT; ++j) {
        int i = base + j * U_THREADS;
        float dx = px[i] - sx, dy = py[i] - sy, dz = pz[i] - sz;
        float d = dx * dx + dy * dy + dz * dz;
        float old = mind[i];
        float md = old;
        if (d < old) { mind[i] = d; md = d; }   // -inf (invalid) never updates
        lbest = umax64(lbest, packKey(md, i));
    }
    sdata[tid] = lbest;
    __syncthreads();
    for (int s = U_THREADS / 2; s > 0; s >>= 1) {
        if (tid < s) sdata[tid] = umax64(sdata[tid], sdata[tid + s]);
        __syncthreads();
    }
    if (tid == 0) pout[blockIdx.x] = sdata[0];
}

// gather [512][8] padded features + build zero-padded W1 [8][64]
__global__ void kgather(const float* __restrict__ x, const int* __restrict__ idxs,
                        const float* __restrict__ px, const float* __restrict__ py,
                        const float* __restrict__ pz, const float* __restrict__ W1,
                        float* __restrict__ feat, float* __restrict__ W1p) {
    int t = blockIdx.x * 512 + threadIdx.x;
    if (t < 512) {
        int idx = idxs[t];
        float* f = feat + t * 8;
        f[0] = px[idx]; f[1] = py[idx]; f[2] = pz[idx];
        const float* xp = x + (size_t)idx * 5;
        const float inv255 = 1.0f / 255.0f;
        f[3] = xp[0] * inv255; f[4] = xp[1] * inv255; f[5] = xp[2] * inv255;
        f[6] = 0.0f; f[7] = 0.0f;
    } else {
        int j = t - 512;                 // 0..511 -> [8][64]
        int kk = j >> 6, n = j & 63;
        W1p[j] = (kk < 6) ? W1[kk * 64 + n] : 0.0f;
    }
}

// one MLP layer: Hout[512,N] = act(LN(Hin[512,K] @ W[K,N] + b))
// 32 blocks of 16 rows; 4 waves per block; f32 WMMA 16x16x4.
__global__ __launch_bounds__(128) void kmlp(
        const float* __restrict__ Hin, const float* __restrict__ W,
        const float* __restrict__ bias, const float* __restrict__ gamma,
        const float* __restrict__ beta, float* __restrict__ Hout,
        int Kdim, int Ndim, int nlog2, int doLN) {
    __shared__ float tile[16 * 512];
    __shared__ float sm[16], sv[16];
    int tid  = threadIdx.x;
    int wid  = tid >> 5;
    int lane = tid & 31;
    int m0   = blockIdx.x * 16;
    int lr   = lane & 15;
    int hk   = (lane >> 4) << 1;        // 0 for lanes 0-15, 2 for lanes 16-31
    int nTiles = Ndim >> 4;

    for (int nt = wid; nt < nTiles; nt += 4) {
        v8f acc = {};
        int cn = nt * 16 + lr;
        for (int kk = 0; kk < Kdim; kk += 4) {
            int kb = kk + hk;
            v2f a, b;
            a.x = Hin[(size_t)(m0 + lr) * Kdim + kb];
            a.y = Hin[(size_t)(m0 + lr) * Kdim + kb + 1];
            b.x = W[(size_t)kb * Ndim + cn];
            b.y = W[(size_t)(kb + 1) * Ndim + cn];
            acc = __builtin_amdgcn_wmma_f32_16x16x4_f32(
                false, a, false, b, (short)0, acc, false, false);
        }
        int rbase = (lane >> 4) << 3;   // rows 0-7 or 8-15
#pragma unroll
        for (int r = 0; r < 8; ++r)
            tile[(r + rbase) * Ndim + cn] = acc[r] + bias[cn];
    }
    __syncthreads();

    if (doLN) {
        if (tid < 16) {
            float s = 0.0f;
            for (int c = 0; c < Ndim; ++c) s += tile[tid * Ndim + c];
            float m = s / (float)Ndim;
            float v = 0.0f;
            for (int c = 0; c < Ndim; ++c) {
                float d = tile[tid * Ndim + c] - m;
                v += d * d;
            }
            sm[tid] = m;
            sv[tid] = rsqrtf(v / (float)Ndim + LN_EPS);
        }
        __syncthreads();
        for (int idx = tid; idx < 16 * Ndim; idx += 128) {
            int row = idx >> nlog2, col = idx & (Ndim - 1);
            float o = (tile[idx] - sm[row]) * sv[row] * gamma[col] + beta[col];
            Hout[(size_t)(m0 + row) * Ndim + col] = fmaxf(o, 0.0f);
        }
    } else {
        for (int idx = tid; idx < 16 * Ndim; idx += 128) {
            int row = idx >> nlog2, col = idx & (Ndim - 1);
            Hout[(size_t)(m0 + row) * Ndim + col] = tile[idx];
        }
    }
}

__global__ void kmaxpool(const float* __restrict__ H4, float* __restrict__ pooled) {
    int col = blockIdx.x * 256 + threadIdx.x;   // 512 columns
    float m = -__builtin_inff();
    for (int r = 0; r < 512; ++r) m = fmaxf(m, H4[(size_t)r * 512 + col]);
    pooled[col] = m;
}

__global__ void kfinal(const float* __restrict__ pooled, const float* __restrict__ Wf,
                       const float* __restrict__ bf, const float* __restrict__ gf,
                       const float* __restrict__ bef, float* __restrict__ out) {
    __shared__ float sy[64];
    int n = threadIdx.x;                        // 64 threads
    float s = bf[n];
    for (int k = 0; k < 512; ++k) s += pooled[k] * Wf[k * 64 + n];
    sy[n] = s;
    __syncthreads();
    float mean = 0.0f;
    for (int i = 0; i < 64; ++i) mean += sy[i];
    mean *= (1.0f / 64.0f);
    float var = 0.0f;
    for (int i = 0; i < 64; ++i) { float d = sy[i] - mean; var += d * d; }
    var *= (1.0f / 64.0f);
    out[n] = (s - mean) * rsqrtf(var + LN_EPS) * gf[n] + bef[n];
}

// ---------------- launch ----------------
extern "C" void kernel_launch(void* const* d_in, const int* in_sizes, int n_in,
                              void* d_out, int out_size, void* d_ws, size_t ws_size,
                              hipStream_t stream) {
    const float* x   = (const float*)d_in[0];
    const float* W1  = (const float*)d_in[1];
    const float* b1  = (const float*)d_in[2];
    const float* W2  = (const float*)d_in[3];
    const float* b2  = (const float*)d_in[4];
    const float* W3  = (const float*)d_in[5];
    const float* b3  = (const float*)d_in[6];
    const float* W4  = (const float*)d_in[7];
    const float* b4  = (const float*)d_in[8];
    const float* g1  = (const float*)d_in[9];
    const float* be1 = (const float*)d_in[10];
    const float* g2  = (const float*)d_in[11];
    const float* be2 = (const float*)d_in[12];
    const float* g3  = (const float*)d_in[13];
    const float* be3 = (const float*)d_in[14];
    const float* Wf  = (const float*)d_in[15];
    const float* bf  = (const float*)d_in[16];
    const float* gf  = (const float*)d_in[17];
    const float* bef = (const float*)d_in[18];

    float* ws   = (float*)d_ws;
    float* px   = ws + OFF_PX;
    float* py   = ws + OFF_PY;
    float* pz   = ws + OFF_PZ;
    float* mind = ws + OFF_MIND;
    int*   ctrl = (int*)(ws + OFF_CTRL);
    unsigned long long* pA = (unsigned long long*)(ws + OFF_PARTA);
    unsigned long long* pB = (unsigned long long*)(ws + OFF_PARTB);
    int*   idxs = (int*)(ws + OFF_IDXS);
    float* feat = ws + OFF_FEAT;
    float* W1p  = ws + OFF_W1P;
    float* H1   = ws + OFF_H1;
    float* H2   = ws + OFF_H2;
    float* H3   = ws + OFF_H3;
    float* H4   = ws + OFF_H4;
    float* pool = ws + OFF_POOL;

    kinit<<<1, 1, 0, stream>>>(ctrl);
    kpoints<<<NPTS / 256, 256, 0, stream>>>(x, px, py, pz, mind, ctrl);
    kseed<<<1, 512, 0, stream>>>(ctrl, pA);

    for (int k = 0; k < KFPS; ++k) {
        unsigned long long* pin  = (k & 1) ? pB : pA;
        unsigned long long* pout = (k & 1) ? pA : pB;
        kfps<<<U_BLOCKS, U_THREADS, 0, stream>>>(px, py, pz, mind, pin, pout, idxs, k);
    }

    kgather<<<2, 512, 0, stream>>>(x, idxs, px, py, pz, W1, feat, W1p);

    kmlp<<<32, 128, 0, stream>>>(feat, W1p, b1, g1, be1, H1,   8,  64, 6, 1);
    kmlp<<<32, 128, 0, stream>>>(H1,   W2,  b2, g2, be2, H2,  64, 128, 7, 1);
    kmlp<<<32, 128, 0, stream>>>(H2,   W3,  b3, g3, be3, H3, 128, 256, 8, 1);
    kmlp<<<32, 128, 0, stream>>>(H3,   W4,  b4, b4, b4,  H4, 256, 512, 9, 0);

    kmaxpool<<<2, 256, 0, stream>>>(H4, pool);
    kfinal<<<1, 64, 0, stream>>>(pool, Wf, bf, gf, bef, (float*)d_out);
}